// PyTorchUnscentedKalmanFilter_28389733826578
// MI455X (gfx1250) — compile-verified
//
#include <hip/hip_runtime.h>
#include <math.h>

// ---------------------------------------------------------------------------
// Batched UKF for MI455X (gfx1250, wave32).
// One workgroup (256 threads = 8 waves) per batch element, everything in LDS.
// Dense GEMM chain runs on V_WMMA_F32_16X16X4_F32 (fp32 matrix pipe);
// Cholesky / triangular inverse run on VALU with workgroup parallelism.
// ---------------------------------------------------------------------------

typedef float v2f __attribute__((ext_vector_type(2)));
typedef float v8f __attribute__((ext_vector_type(8)));

#define DIM_X 64
#define DIM_Z 32
#define LDL   65      // row stride (floats) for 64-wide LDS tiles (bank-conflict pad)
#define LDK   33      // row stride for 32-wide LDS tiles
#define QN    0.01f   // process noise
#define RN    0.1f    // measurement noise
#define OUT_PER_B (DIM_X + DIM_X * DIM_X)  // 4160

// A-fragment (and B=Xᵀ fragment) for 16x16x4 f32 WMMA, row-major source:
// lane 0-15 -> M = lane, K = k,k+1 ; lane 16-31 -> same M rows, K = k+2,k+3
__device__ __forceinline__ v2f frag_row(const float* m, int ld, int row0, int k, int lane) {
  int r  = lane & 15;
  int kk = k + ((lane >> 4) << 1);
  const float* p = m + (row0 + r) * ld + kk;
  v2f a; a.x = p[0]; a.y = p[1];
  return a;
}

// B-fragment for B[k][n] = X[k][col0+n] (X stored row-major, row index is K)
__device__ __forceinline__ v2f frag_col(const float* m, int ld, int col0, int k, int lane) {
  int r  = lane & 15;
  int kk = k + ((lane >> 4) << 1);
  v2f a;
  a.x = m[kk * ld + col0 + r];
  a.y = m[(kk + 1) * ld + col0 + r];
  return a;
}

__device__ __forceinline__ v8f wmma4(v2f a, v2f b, v8f c) {
  // 8-arg form: (neg_a, A, neg_b, B, c_mod, C, reuse_a, reuse_b)
  return __builtin_amdgcn_wmma_f32_16x16x4_f32(false, a, false, b, (short)0, c, false, false);
}

// D layout: VGPR i -> row = row0 + i + 8*(lane>=16), col = col0 + (lane&15)
__device__ __forceinline__ void store_tile(float* m, int ld, int row0, int col0, int lane, v8f c) {
  int n     = col0 + (lane & 15);
  int rbase = row0 + ((lane >> 4) << 3);
#pragma unroll
  for (int i = 0; i < 8; ++i) m[(rbase + i) * ld + n] = c[i];
}

__global__ __launch_bounds__(256)
void ukf_kernel(const float* __restrict__ x, const float* __restrict__ P,
                const float* __restrict__ z, float* __restrict__ out) {
  __shared__ float Ls[DIM_X * LDL];   // P -> chol(P) = L
  __shared__ float Mm[DIM_X * LDL];   // M = L*Lᵀ
  __shared__ float Kt[DIM_X * LDK];   // Kalman gain K
  __shared__ float Tt[DIM_X * LDK];   // T = K*S
  __shared__ float Sc[DIM_Z * LDK];   // chol(S)
  __shared__ float Li[DIM_Z * LDK];   // inv(chol(S)) (lower)
  __shared__ float Si[DIM_Z * LDK];   // S^-1
  __shared__ float dz[DIM_Z];

  const int b    = blockIdx.x;
  const int tid  = threadIdx.x;
  const int lane = tid & 31;
  const int wave = tid >> 5;

  const float* Pb = P + (size_t)b * DIM_X * DIM_X;
  const float* xb = x + (size_t)b * DIM_X;
  const float* zb = z + (size_t)b * DIM_Z;
  float*       ob = out + (size_t)b * OUT_PER_B;

  // ---- load P into LDS ----
  for (int i = tid; i < DIM_X * DIM_X; i += 256)
    Ls[(i >> 6) * LDL + (i & 63)] = Pb[i];
  if (tid < DIM_Z) dz[tid] = zb[tid] - xb[tid];   // innovation (z_pred = x[:32])
  __syncthreads();

  // ---- Cholesky of P (right-looking; rectangle update keeps symmetry) ----
  for (int j = 0; j < DIM_X; ++j) {
    if (tid == 0) Ls[j * LDL + j] = sqrtf(Ls[j * LDL + j]);
    __syncthreads();
    float dinv = 1.0f / Ls[j * LDL + j];
    for (int i = j + 1 + tid; i < DIM_X; i += 256) Ls[i * LDL + j] *= dinv;
    __syncthreads();
    int m = DIM_X - 1 - j;
    for (int idx = tid; idx < m * m; idx += 256) {
      int i = j + 1 + idx / m;
      int k = j + 1 + idx % m;
      Ls[i * LDL + k] -= Ls[i * LDL + j] * Ls[k * LDL + j];
    }
    __syncthreads();
  }
  for (int i = tid; i < DIM_X * DIM_X; i += 256) {   // zero strict upper
    int r = i >> 6, c = i & 63;
    if (c > r) Ls[r * LDL + c] = 0.0f;
  }
  __syncthreads();

  // ---- M = L*Lᵀ  (UKF weights make P_pred = M + q·I exactly) : 16 tiles ----
  for (int t = wave; t < 16; t += 8) {
    int ti = t >> 2, tj = t & 3;
    v8f acc = {};
#pragma unroll
    for (int k = 0; k < DIM_X; k += 4) {
      v2f a  = frag_row(Ls, LDL, ti * 16, k, lane);
      v2f bf = frag_row(Ls, LDL, tj * 16, k, lane);   // B = Lᵀ
      acc = wmma4(a, bf, acc);
    }
    store_tile(Mm, LDL, ti * 16, tj * 16, lane, acc);
  }
  __syncthreads();

  // ---- S = M[:32,:32] + r·I, then chol(S) ----
  for (int i = tid; i < DIM_Z * DIM_Z; i += 256) {
    int r = i >> 5, c = i & 31;
    Sc[r * LDK + c] = Mm[r * LDL + c] + ((r == c) ? RN : 0.0f);
  }
  __syncthreads();
  for (int j = 0; j < DIM_Z; ++j) {
    if (tid == 0) Sc[j * LDK + j] = sqrtf(Sc[j * LDK + j]);
    __syncthreads();
    float dinv = 1.0f / Sc[j * LDK + j];
    for (int i = j + 1 + tid; i < DIM_Z; i += 256) Sc[i * LDK + j] *= dinv;
    __syncthreads();
    int m = DIM_Z - 1 - j;
    for (int idx = tid; idx < m * m; idx += 256) {
      int i = j + 1 + idx / m;
      int k = j + 1 + idx % m;
      Sc[i * LDK + k] -= Sc[i * LDK + j] * Sc[k * LDK + j];
    }
    __syncthreads();
  }

  // ---- Li = inv(chol(S)), forward substitution, one column per lane ----
  if (tid < DIM_Z) {
    int j = tid;
    for (int i = 0; i < DIM_Z; ++i) {
      float s;
      if (i < j) s = 0.0f;
      else {
        s = (i == j) ? 1.0f : 0.0f;
        for (int k = j; k < i; ++k) s -= Sc[i * LDK + k] * Li[k * LDK + j];
        s /= Sc[i * LDK + i];
      }
      Li[i * LDK + j] = s;
    }
  }
  __syncthreads();

  // ---- Si = Liᵀ*Li  (4 tiles on waves 0..3) ----
  if (wave < 4) {
    int ti = wave >> 1, tj = wave & 1;
    v8f acc = {};
#pragma unroll
    for (int k = 0; k < DIM_Z; k += 4) {
      v2f a  = frag_col(Li, LDK, ti * 16, k, lane);   // A = Liᵀ
      v2f bf = frag_col(Li, LDK, tj * 16, k, lane);
      acc = wmma4(a, bf, acc);
    }
    store_tile(Si, LDK, ti * 16, tj * 16, lane, acc);
  }
  __syncthreads();

  // ---- K = Pxz * S^-1  (Pxz = M[:, :32]) : 8 tiles, one per wave ----
  {
    int ti = wave >> 1, tj = wave & 1;
    v8f acc = {};
#pragma unroll
    for (int k = 0; k < DIM_Z; k += 4) {
      v2f a  = frag_row(Mm, LDL, ti * 16, k, lane);
      v2f bf = frag_col(Si, LDK, tj * 16, k, lane);
      acc = wmma4(a, bf, acc);
    }
    store_tile(Kt, LDK, ti * 16, tj * 16, lane, acc);
  }
  __syncthreads();

  // ---- T = K*S, reading S on the fly as M[:32,:32] + r·I : 8 tiles ----
  {
    int ti = wave >> 1, tj = wave & 1;
    v8f acc = {};
#pragma unroll
    for (int k = 0; k < DIM_Z; k += 4) {
      v2f a = frag_row(Kt, LDK, ti * 16, k, lane);
      int r  = lane & 15;
      int kk = k + ((lane >> 4) << 1);
      int c0 = tj * 16 + r;
      v2f bf;
      bf.x = Mm[kk * LDL + c0]       + ((kk == c0)       ? RN : 0.0f);
      bf.y = Mm[(kk + 1) * LDL + c0] + (((kk + 1) == c0) ? RN : 0.0f);
      acc = wmma4(a, bf, acc);
    }
    store_tile(Tt, LDK, ti * 16, tj * 16, lane, acc);
  }
  __syncthreads();

  // ---- x_new = x + K*dz ----
  if (tid < DIM_X) {
    float s = xb[tid];
    for (int j = 0; j < DIM_Z; ++j) s += Kt[tid * LDK + j] * dz[j];
    ob[tid] = s;
  }

  // ---- P_new = (M + q·I) - T*Kᵀ, written straight to d_out : 16 tiles ----
  for (int t = wave; t < 16; t += 8) {
    int ti = t >> 2, tj = t & 3;
    v8f acc = {};
#pragma unroll
    for (int k = 0; k < DIM_Z; k += 4) {
      v2f a  = frag_row(Tt, LDK, ti * 16, k, lane);
      v2f bf = frag_row(Kt, LDK, tj * 16, k, lane);   // B = Kᵀ
      acc = wmma4(a, bf, acc);
    }
    int n     = tj * 16 + (lane & 15);
    int rbase = ti * 16 + ((lane >> 4) << 3);
#pragma unroll
    for (int i = 0; i < 8; ++i) {
      int row = rbase + i;
      ob[DIM_X + row * DIM_X + n] =
          Mm[row * LDL + n] + ((row == n) ? QN : 0.0f) - acc[i];
    }
  }
}

extern "C" void kernel_launch(void* const* d_in, const int* in_sizes, int n_in,
                              void* d_out, int out_size, void* d_ws, size_t ws_size,
                              hipStream_t stream) {
  (void)n_in; (void)d_ws; (void)ws_size; (void)out_size;
  const float* x = (const float*)d_in[0];   // (B, 64)
  const float* P = (const float*)d_in[1];   // (B, 64, 64)
  const float* z = (const float*)d_in[2];   // (B, 32)
  float* out = (float*)d_out;               // (B, 4160)
  int B = in_sizes[0] / DIM_X;
  hipLaunchKernelGGL(ukf_kernel, dim3(B), dim3(256), 0, stream, x, P, z, out);
}